// DeconvLayer_43052752175212
// MI455X (gfx1250) — compile-verified
//
#include <hip/hip_runtime.h>

typedef float v2f __attribute__((ext_vector_type(2)));
typedef float v4f __attribute__((ext_vector_type(4)));
typedef float v8f __attribute__((ext_vector_type(8)));

#define N_LEN 131072
#define F_TAPS 16
// tiles of 16 positions covering [32, N): (131072-32)/16 = 8190 tiles
#define TILES_TOTAL 8190
#define STRIP_TILES 30          // 8190 = 273 * 30, multiple of 3 for the unroll
#define STRIPS_PER_GROUP 273
#define GROUPS 16               // 256 batch rows / 16
#define WAVES_PER_BLOCK 8
#define NUM_BLOCKS ((GROUPS * STRIPS_PER_GROUP) / WAVES_PER_BLOCK)  // 546

__device__ __forceinline__ float k1_tap(const float* h, int a) {
    return (a == 0) ? 1.0f : h[a - 1];          // k1 = [1, h[0..15]]
}
__device__ __forceinline__ float k2_tap(const float* h, int b) {
    return (b == 0) ? 1.0f : h[16 - b];         // k2 = [1, h[15], ..., h[0]]
}

// One 16x16 output tile: D = T2*Xold + T1*Xmid + T0*Xnew, then store 8 floats/lane.
__device__ __forceinline__ void tile_step(const v2f a[3][4],
                                          const v2f* bo, const v2f* bm, const v2f* bn,
                                          float* yptr) {
    v8f acc = {};
#pragma unroll
    for (int c = 0; c < 4; ++c)
        acc = __builtin_amdgcn_wmma_f32_16x16x4_f32(false, a[2][c], false, bo[c],
                                                    (short)0, acc, false, false);
#pragma unroll
    for (int c = 0; c < 4; ++c)
        acc = __builtin_amdgcn_wmma_f32_16x16x4_f32(false, a[1][c], false, bm[c],
                                                    (short)0, acc, false, false);
#pragma unroll
    for (int c = 0; c < 4; ++c)
        acc = __builtin_amdgcn_wmma_f32_16x16x4_f32(false, a[0][c], false, bn[c],
                                                    (short)0, acc, false, false);
    v4f lo = {acc[0], acc[1], acc[2], acc[3]};
    v4f hi = {acc[4], acc[5], acc[6], acc[7]};
    *(v4f*)(yptr)     = lo;
    *(v4f*)(yptr + 4) = hi;
}

__global__ void __launch_bounds__(256) fir_wmma(const float* __restrict__ x,
                                                const float* __restrict__ h,
                                                float* __restrict__ y) {
    // Build padded 33-tap combined kernel Ktot = k2 (*) k1 in LDS, with 15 zeros
    // of left pad and zeros on the right: sKpad[15 + t] = Ktot[t], t in [0,32].
    __shared__ float sKpad[64];
    const int tid = threadIdx.x;
    if (tid < 63) {
        const int t = tid - 15;
        float v = 0.0f;
        if (t >= 0 && t <= 32) {
            const int alo = (t > 16) ? (t - 16) : 0;
            const int ahi = (t < 16) ? t : 16;
            float s = 0.0f;
            for (int a = alo; a <= ahi; ++a) s += k1_tap(h, a) * k2_tap(h, t - a);
            v = s;
        }
        sKpad[tid] = v;
    }
    __syncthreads();

    const int wave = tid >> 5;
    const int lane = tid & 31;
    const int lo16 = lane & 15;
    const int half = lane >> 4;

    const int wave_id = blockIdx.x * WAVES_PER_BLOCK + wave;   // 0..4367
    const int group   = wave_id / STRIPS_PER_GROUP;            // batch group
    const int strip   = wave_id % STRIPS_PER_GROUP;
    const int b0      = group * 16;
    const int P0      = 32 + strip * (STRIP_TILES * 16);

    // A fragments: Toeplitz chunks of Ktot. Lane layout for f32 16x4 A:
    // M = lane%16, K_local = 2*(lane/16) + component.
    v2f a[3][4];
    const int m = lo16;
#pragma unroll
    for (int blk = 0; blk < 3; ++blk) {
#pragma unroll
        for (int c = 0; c < 4; ++c) {
            const int p0  = 4 * c + 2 * half;
            const int idx = 15 + m - p0 + 16 * blk;   // 0..62, always in range
            a[blk][c].x = sKpad[idx];
            a[blk][c].y = sKpad[idx - 1];
        }
    }

    const float* xrow = x + (size_t)(b0 + lo16) * N_LEN;
    float*       yrow = y + (size_t)(b0 + lo16) * N_LEN;

    // Rolling B fragments (f32 4x16 B layout: N = lane%16, K_local = 2*(lane/16)+e),
    // so each lane loads 2 consecutive floats of its own batch row.
    v2f B0[4], B1[4], B2[4];
#define LOADB(dst, pos)                                            \
    {                                                              \
        _Pragma("unroll")                                          \
        for (int c = 0; c < 4; ++c)                                \
            dst[c] = *(const v2f*)(xrow + (pos) + 4 * c + 2 * half); \
    }

    LOADB(B0, P0 - 32);
    LOADB(B1, P0 - 16);

    for (int tt = 0; tt < STRIP_TILES; tt += 3) {
        const int P = P0 + 16 * tt;
        __builtin_prefetch(xrow + P + 96, 0, 0);   // global_prefetch_b8

        LOADB(B2, P);
        tile_step(a, B0, B1, B2, yrow + P + 8 * half);

        LOADB(B0, P + 16);
        tile_step(a, B1, B2, B0, yrow + P + 16 + 8 * half);

        LOADB(B1, P + 32);
        tile_step(a, B2, B0, B1, yrow + P + 32 + 8 * half);
    }
#undef LOADB
}

// Exact two-pass (masked) evaluation for positions [0, 32) of every row.
__global__ void fir_boundary(const float* __restrict__ x,
                             const float* __restrict__ h,
                             float* __restrict__ y) {
    const int b = blockIdx.x;       // 256 rows
    const int i = threadIdx.x;      // 32 positions
    const float* xr = x + (size_t)b * N_LEN;
    float out = 0.0f;
    if (i >= 16) {
        float hl[16];
#pragma unroll
        for (int j = 0; j < 16; ++j) hl[j] = h[j];
        // v(k) = x[k] + sum_j h[j] * x[k-1-j], valid for k >= 16
        float s;
        {
            float acc = xr[i];
#pragma unroll
            for (int j = 0; j < 16; ++j) acc += hl[j] * xr[i - 1 - j];
            s = acc;
        }
        for (int j = 0; j <= i - 17; ++j) {       // only v indices >= 16 survive
            const int k = i - 1 - j;
            float acc = xr[k];
#pragma unroll
            for (int jj = 0; jj < 16; ++jj) acc += hl[jj] * xr[k - 1 - jj];
            s += hl[15 - j] * acc;                 // hr[j] = h[15-j]
        }
        out = s;
    }
    y[(size_t)b * N_LEN + i] = out;
}

extern "C" void kernel_launch(void* const* d_in, const int* in_sizes, int n_in,
                              void* d_out, int out_size, void* d_ws, size_t ws_size,
                              hipStream_t stream) {
    const float* x = (const float*)d_in[0];   // (256, 131072) f32
    const float* h = (const float*)d_in[1];   // (1, 16) f32
    float* y = (float*)d_out;                 // (256, 131072) f32
    (void)in_sizes; (void)n_in; (void)out_size; (void)d_ws; (void)ws_size;

    fir_boundary<<<256, 32, 0, stream>>>(x, h, y);
    fir_wmma<<<NUM_BLOCKS, 256, 0, stream>>>(x, h, y);
}